// AttentionEncoder_68427418960337
// MI455X (gfx1250) — compile-verified
//
#include <hip/hip_runtime.h>
#include <hip/hip_bf16.h>

// ---------------------------------------------------------------------------
// AttentionEncoder on MI455X (gfx1250): memory-bound -> stay in f32 and use
// V_WMMA_F32_16X16X4_F32 for every contraction (qkv GEMM, Gram, attn*V, proj).
// ---------------------------------------------------------------------------

typedef float v2f __attribute__((ext_vector_type(2)));
typedef float v8f __attribute__((ext_vector_type(8)));

#define NB 8
#define CDIM 128
#define C3 384
#define HEADS 8
#define HW 16384      // 128*128
#define WPIX 128

__device__ __forceinline__ v8f wmma_f32(v2f a, v2f b, v8f c) {
    // 8 args: (neg_a, A, neg_b, B, c_mod, C, reuse_a, reuse_b)
    return __builtin_amdgcn_wmma_f32_16x16x4_f32(false, a, false, b,
                                                 (short)0, c, false, false);
}

// ---------------------------------------------------------------------------
// Y[b, m, n] = sum_c W[m, c] * X[b, c, n]   (K = 128, N = 16384)
// One wave computes one 16x16 output tile; K-loop of 32 WMMA(16x16x4) steps.
// ---------------------------------------------------------------------------
__global__ void gemm1x1_kernel(const float* __restrict__ W,   // [M,128]
                               const float* __restrict__ X,   // [B,128,N]
                               float* __restrict__ Y,         // [B,M,N]
                               int M) {
    const int K = CDIM;
    const int lane = threadIdx.x & 31;
    const int wave = threadIdx.x >> 5;

    const long tile     = (long)blockIdx.x * 8 + wave;
    const int  ntiles_n = HW / 16;           // 1024
    const int  ntiles_m = M / 16;
    const long tiles_b  = (long)ntiles_m * ntiles_n;
    const int  b  = (int)(tile / tiles_b);
    const long t  = tile % tiles_b;
    const int  mt = (int)(t / ntiles_n);
    const int  nt = (int)(t % ntiles_n);

    const float* Wp = W + (long)(mt * 16) * K;
    const float* Xp = X + (long)b * K * HW + nt * 16;
    float*       Yp = Y + (long)b * M * HW + (long)(mt * 16) * HW + nt * 16;

    const int m      = lane & 15;            // A row (M)
    const int ncol   = lane & 15;            // B/D column (N)
    const int khalf2 = (lane >> 4) << 1;     // lanes 0-15 -> K{0,1}, 16-31 -> K{2,3}

    __builtin_prefetch(Wp + (long)m * K, 0, 1);   // global_prefetch_b8

    v8f acc = {};
    for (int k = 0; k < K; k += 4) {
        v2f a, bb;
        a.x  = Wp[(long)m * K + k + khalf2];
        a.y  = Wp[(long)m * K + k + khalf2 + 1];
        bb.x = Xp[(long)(k + khalf2) * HW + ncol];
        bb.y = Xp[(long)(k + khalf2 + 1) * HW + ncol];
        acc  = wmma_f32(a, bb, acc);
    }

    const int mbase = (lane >> 4) * 8;
#pragma unroll
    for (int r = 0; r < 8; ++r)
        Yp[(long)(mbase + r) * HW + ncol] = acc[r];
}

// ---------------------------------------------------------------------------
// Depthwise 3x3, SAME padding (cross-correlation, matching lax.conv).
// ---------------------------------------------------------------------------
__global__ void dwconv3_kernel(const float* __restrict__ X,   // [B,384,128,128]
                               const float* __restrict__ Wd,  // [384,1,3,3]
                               float* __restrict__ Y) {
    const long idx = (long)blockIdx.x * blockDim.x + threadIdx.x;
    const int  x   = (int)(idx & 127);
    const int  y   = (int)((idx >> 7) & 127);
    const long cb  = idx >> 14;              // b*384 + ch
    const int  ch  = (int)(cb % C3);
    const float* Xc = X + (cb << 14);
    const float* wp = Wd + (long)ch * 9;

    float s = 0.f;
#pragma unroll
    for (int ky = 0; ky < 3; ++ky) {
        const int yy = y + ky - 1;
        if (yy < 0 || yy >= WPIX) continue;
#pragma unroll
        for (int kx = 0; kx < 3; ++kx) {
            const int xx = x + kx - 1;
            if (xx < 0 || xx >= WPIX) continue;
            s += Xc[(long)yy * WPIX + xx] * wp[ky * 3 + kx];
        }
    }
    Y[idx] = s;
}

// ---------------------------------------------------------------------------
// Reciprocal L2 norms over the 16384-pixel axis.
// grid.y selects the set: 0=q_img, 1=k_img, 2=q_edge, 3=k_edge
// grid.x = B*128 rows; block = 256.
// ---------------------------------------------------------------------------
__global__ void invnorm_kernel(const float* __restrict__ dwi,
                               const float* __restrict__ dwe,
                               float* __restrict__ inrm) {     // [4][B*128]
    const int set = blockIdx.y;
    const int row = blockIdx.x;              // b*128 + c
    const int b = row >> 7, c = row & 127;
    const float* base  = (set < 2) ? dwi : dwe;
    const int    choff = (set & 1) ? CDIM : 0;
    const float* p = base + ((long)b * C3 + choff + c) * HW;

    float s = 0.f;
    for (int i = threadIdx.x; i < HW; i += 256) {
        const float v = p[i];
        s += v * v;
    }
    __shared__ float red[256];
    red[threadIdx.x] = s;
    __syncthreads();
    for (int st = 128; st > 0; st >>= 1) {
        if ((int)threadIdx.x < st) red[threadIdx.x] += red[threadIdx.x + st];
        __syncthreads();
    }
    if (threadIdx.x == 0) {
        const float n = fmaxf(sqrtf(red[0]), 1e-12f);
        inrm[set * (NB * CDIM) + row] = 1.0f / n;
    }
}

// ---------------------------------------------------------------------------
// Channel-attention Gram + softmax.
// grid = (64 = b*head, 2 = stream), block = 256 (8 waves).
// Wave w accumulates S[16x16] over pixels [w*2048, (w+1)*2048) with WMMA,
// with l2-normalization and k = alpha1*k_e_hat + k_hat fused into the loads.
// ---------------------------------------------------------------------------
__global__ void gram_softmax_kernel(const float* __restrict__ dwi,
                                    const float* __restrict__ dwe,
                                    const float* __restrict__ inrm,
                                    const float* __restrict__ temp,   // [8]
                                    const float* __restrict__ tempe,  // [8]
                                    const float* __restrict__ alpha1p,
                                    float* __restrict__ attn) {       // [2][64][16][16]
    const int bh     = blockIdx.x;           // b*8 + head
    const int stream = blockIdx.y;           // 0 = img, 1 = edge
    const int b = bh >> 3, head = bh & 7;
    const int lane = threadIdx.x & 31, wave = threadIdx.x >> 5;
    const float alpha1 = *alpha1p;

    const int cbase = head * 16;
    const float* qs     = ((stream == 0) ? dwi : dwe) + ((long)b * C3 + cbase) * HW;
    const float* ks_img = dwi + ((long)b * C3 + CDIM + cbase) * HW;
    const float* ks_edg = dwe + ((long)b * C3 + CDIM + cbase) * HW;

    const int m      = lane & 15;            // A row  (q channel)
    const int d      = lane & 15;            // B col  (k channel)
    const int khalf2 = (lane >> 4) << 1;

    const int qset = (stream == 0) ? 0 : 2;
    const float sq   = inrm[qset * (NB * CDIM) + b * CDIM + cbase + m];
    const float nk_i = inrm[1 * (NB * CDIM) + b * CDIM + cbase + d];
    const float nk_e = inrm[3 * (NB * CDIM) + b * CDIM + cbase + d];

    const float* qrow = qs     + (long)m * HW;
    const float* kir  = ks_img + (long)d * HW;
    const float* ker  = ks_edg + (long)d * HW;

    v8f acc = {};
    const int n0 = wave * 2048;
    for (int n = n0; n < n0 + 2048; n += 4) {
        v2f a, bb;
        a.x = qrow[n + khalf2]     * sq;
        a.y = qrow[n + khalf2 + 1] * sq;
        if (stream == 0) {
            bb.x = kir[n + khalf2]     * nk_i + alpha1 * ker[n + khalf2]     * nk_e;
            bb.y = kir[n + khalf2 + 1] * nk_i + alpha1 * ker[n + khalf2 + 1] * nk_e;
        } else {
            bb.x = ker[n + khalf2]     * nk_e;
            bb.y = ker[n + khalf2 + 1] * nk_e;
        }
        acc = wmma_f32(a, bb, acc);
    }

    __shared__ float part[8][16][16];
    const int mbase = (lane >> 4) * 8;
#pragma unroll
    for (int r = 0; r < 8; ++r) part[wave][mbase + r][d] = acc[r];
    __syncthreads();

    __shared__ float S[16][17];
    {
        const int row = threadIdx.x >> 4, col = threadIdx.x & 15;
        float s = 0.f;
#pragma unroll
        for (int wv = 0; wv < 8; ++wv) s += part[wv][row][col];
        const float tt = (stream == 0) ? temp[head] : tempe[head];
        S[row][col] = s * tt;
    }
    __syncthreads();

    if (threadIdx.x < 16) {
        const int row = threadIdx.x;
        float mx = -1e30f;
        for (int c2 = 0; c2 < 16; ++c2) mx = fmaxf(mx, S[row][c2]);
        float e[16], sum = 0.f;
        for (int c2 = 0; c2 < 16; ++c2) { e[c2] = expf(S[row][c2] - mx); sum += e[c2]; }
        float* out = attn + ((long)stream * 64 + bh) * 256 + row * 16;
        const float inv = 1.0f / sum;
        for (int c2 = 0; c2 < 16; ++c2) out[c2] = e[c2] * inv;
    }
}

// ---------------------------------------------------------------------------
// out[16c x 16n] = attn[16x16] @ v_mix[16 x 16n], v_mix = v + alpha2*v_e.
// grid = (128 n-tile groups, 64 b*head, 2 streams), block = 256 (8 waves).
// ---------------------------------------------------------------------------
__global__ void av_kernel(const float* __restrict__ dwi,
                          const float* __restrict__ dwe,
                          const float* __restrict__ attn,
                          const float* __restrict__ alpha2p,
                          float* __restrict__ out_img,     // [B,128,HW]
                          float* __restrict__ out_edge) {  // [B,128,HW]
    const int stream = blockIdx.z;
    const int bh = blockIdx.y;
    const int b = bh >> 3, head = bh & 7;
    const int wave = threadIdx.x >> 5, lane = threadIdx.x & 31;
    const int nt = blockIdx.x * 8 + wave;    // 0..1023
    const int n0 = nt * 16;
    const float alpha2 = *alpha2p;

    const float* A  = attn + ((long)stream * 64 + bh) * 256;          // [16][16]
    const float* vi = dwi + ((long)b * C3 + 2 * CDIM + head * 16) * HW;
    const float* ve = dwe + ((long)b * C3 + 2 * CDIM + head * 16) * HW;

    const int m      = lane & 15;
    const int n      = lane & 15;
    const int khalf2 = (lane >> 4) << 1;

    v8f acc = {};
#pragma unroll
    for (int kc = 0; kc < 16; kc += 4) {
        v2f a, bb;
        a.x = A[m * 16 + kc + khalf2];
        a.y = A[m * 16 + kc + khalf2 + 1];
        const int d0 = kc + khalf2;
        if (stream == 0) {
            bb.x = vi[(long)d0 * HW + n0 + n]       + alpha2 * ve[(long)d0 * HW + n0 + n];
            bb.y = vi[(long)(d0 + 1) * HW + n0 + n] + alpha2 * ve[(long)(d0 + 1) * HW + n0 + n];
        } else {
            bb.x = ve[(long)d0 * HW + n0 + n];
            bb.y = ve[(long)(d0 + 1) * HW + n0 + n];
        }
        acc = wmma_f32(a, bb, acc);
    }

    float* out = ((stream == 0) ? out_img : out_edge) +
                 ((long)b * CDIM + head * 16) * HW + n0;
    const int mbase = (lane >> 4) * 8;
#pragma unroll
    for (int r = 0; r < 8; ++r) out[(long)(mbase + r) * HW + n] = acc[r];
}

// ---------------------------------------------------------------------------
// Launch: stage through d_ws (~738 MB of f32 scratch).
// ---------------------------------------------------------------------------
extern "C" void kernel_launch(void* const* d_in, const int* in_sizes, int n_in,
                              void* d_out, int out_size, void* d_ws, size_t ws_size,
                              hipStream_t stream) {
    (void)in_sizes; (void)n_in; (void)out_size; (void)ws_size;

    const float* inp_img   = (const float*)d_in[0];
    const float* inp_edge  = (const float*)d_in[1];
    const float* w_qkv     = (const float*)d_in[2];
    const float* w_dw      = (const float*)d_in[3];
    const float* w_qkv_e   = (const float*)d_in[4];
    const float* w_dw_e    = (const float*)d_in[5];
    const float* w_proj    = (const float*)d_in[6];
    const float* w_proj_e  = (const float*)d_in[7];
    const float* temp      = (const float*)d_in[8];
    const float* temp_e    = (const float*)d_in[9];
    const float* alpha1    = (const float*)d_in[10];
    const float* alpha2    = (const float*)d_in[11];

    const long QKV = (long)NB * C3 * HW;     // 50,331,648 floats
    const long OUT = (long)NB * CDIM * HW;   // 16,777,216 floats

    float* ws    = (float*)d_ws;
    float* raw   = ws;                 // qkv pre-dwconv (reused for both streams)
    float* dwi   = raw   + QKV;        // dwconv(qkv) image
    float* dwe   = dwi   + QKV;        // dwconv(qkv) edge
    float* outa  = dwe   + QKV;        // attn output, image stream
    float* outae = outa  + OUT;        // attn output, edge stream
    float* attn  = outae + OUT;        // [2][64][16][16]
    float* norms = attn  + 2L * 64 * 256; // [4][B*128]

    const int dwBlocks   = (int)(QKV / 256);                       // 196608
    const int qkvBlocks  = (int)((long)NB * (C3 / 16) * (HW / 16) / 8);   // 24576
    const int projBlocks = (int)((long)NB * (CDIM / 16) * (HW / 16) / 8); // 8192

    // 1) qkv = dwconv3(conv1x1(img))
    gemm1x1_kernel<<<qkvBlocks, 256, 0, stream>>>(w_qkv, inp_img, raw, C3);
    dwconv3_kernel<<<dwBlocks, 256, 0, stream>>>(raw, w_dw, dwi);

    // 2) qkv_e = dwconv3(conv1x1(edge))
    gemm1x1_kernel<<<qkvBlocks, 256, 0, stream>>>(w_qkv_e, inp_edge, raw, C3);
    dwconv3_kernel<<<dwBlocks, 256, 0, stream>>>(raw, w_dw_e, dwe);

    // 3) reciprocal L2 norms for q/k of both streams
    invnorm_kernel<<<dim3(NB * CDIM, 4), 256, 0, stream>>>(dwi, dwe, norms);

    // 4) 16x16 channel attention (Gram + softmax), both streams
    gram_softmax_kernel<<<dim3(NB * HEADS, 2), 256, 0, stream>>>(
        dwi, dwe, norms, temp, temp_e, alpha1, attn);

    // 5) out = attn @ v_mix, both streams
    av_kernel<<<dim3(HW / 16 / 8, NB * HEADS, 2), 256, 0, stream>>>(
        dwi, dwe, attn, alpha2, outa, outae);

    // 6) projections straight into d_out (out || out_e)
    float* dout = (float*)d_out;
    gemm1x1_kernel<<<projBlocks, 256, 0, stream>>>(w_proj, outa, dout, CDIM);
    gemm1x1_kernel<<<projBlocks, 256, 0, stream>>>(w_proj_e, outae, dout + OUT, CDIM);
}